// DentalResPointNet_16063177687270
// MI455X (gfx1250) — compile-verified
//
#include <hip/hip_runtime.h>

// ---------------------------------------------------------------------------
// DentalResPointNet forward for MI455X (gfx1250, wave32, WMMA)
// ---------------------------------------------------------------------------

#define NN0 16384
#define NN1 4096
#define NN2 1024
#define NN3 256

typedef __attribute__((ext_vector_type(2))) float v2f;
typedef __attribute__((ext_vector_type(8))) float v8f;

// ---------------- Farthest point sampling (single persistent workgroup) ----
__global__ void fps_kernel(const float* __restrict__ pos, int N, int M,
                           int* __restrict__ idx, float* __restrict__ dist)
{
  __shared__ float smax[256];
  __shared__ int   simax[256];
  const int t = threadIdx.x;
  if (t == 0) idx[0] = 0;
  float lx = pos[0], ly = pos[1], lz = pos[2];
  float bm = -1.0f; int bi = 0;
  for (int j = t; j < N; j += 256) {
    float dx = pos[3*j] - lx, dy = pos[3*j+1] - ly, dz = pos[3*j+2] - lz;
    float d = dx*dx + dy*dy + dz*dz;
    dist[j] = d;
    if (d > bm) { bm = d; bi = j; }
  }
  for (int i = 1; i < M; ++i) {
    smax[t] = bm; simax[t] = bi;
    __syncthreads();
    for (int s = 128; s > 0; s >>= 1) {
      if (t < s && (smax[t+s] > smax[t] ||
                    (smax[t+s] == smax[t] && simax[t+s] < simax[t]))) {
        smax[t] = smax[t+s]; simax[t] = simax[t+s];
      }
      __syncthreads();
    }
    const int nxt = simax[0];
    if (t == 0) idx[i] = nxt;
    lx = pos[3*nxt]; ly = pos[3*nxt+1]; lz = pos[3*nxt+2];
    __syncthreads();
    bm = -1.0f; bi = 0;
    for (int j = t; j < N; j += 256) {
      float dx = pos[3*j]-lx, dy = pos[3*j+1]-ly, dz = pos[3*j+2]-lz;
      float nd = dx*dx + dy*dy + dz*dz;
      float d = fminf(dist[j], nd);
      dist[j] = d;
      if (d > bm) { bm = d; bi = j; }
    }
  }
}

// ---------------- kNN (one workgroup per query; LDS bitmap selection) ------
__global__ void knn_kernel(const float* __restrict__ pos, const int* __restrict__ qidx,
                           int N, int K, int* __restrict__ src)
{
  __shared__ unsigned sel[512];       // bitmap, covers N <= 16384
  __shared__ float rmin[256];
  __shared__ int   rimin[256];
  const int t = threadIdx.x;
  const int q = blockIdx.x;
  const int qi = qidx[q];
  const float qx = pos[3*qi], qy = pos[3*qi+1], qz = pos[3*qi+2];
  for (int j = t; j < 512; j += 256) sel[j] = 0u;
  __syncthreads();
  for (int k = 0; k < K; ++k) {
    float bm = 3.4e38f; int bi = 0x7fffffff;
    for (int j = t; j < N; j += 256) {
      if (sel[j >> 5] & (1u << (j & 31))) continue;
      float dx = pos[3*j]-qx, dy = pos[3*j+1]-qy, dz = pos[3*j+2]-qz;
      float d = dx*dx + dy*dy + dz*dz;
      if (d < bm || (d == bm && j < bi)) { bm = d; bi = j; }
    }
    rmin[t] = bm; rimin[t] = bi;
    __syncthreads();
    for (int s = 128; s > 0; s >>= 1) {
      if (t < s && (rmin[t+s] < rmin[t] ||
                    (rmin[t+s] == rmin[t] && rimin[t+s] < rimin[t]))) {
        rmin[t] = rmin[t+s]; rimin[t] = rimin[t+s];
      }
      __syncthreads();
    }
    if (t == 0) {
      int b = rimin[0];
      src[(size_t)q * K + k] = b;
      sel[b >> 5] |= (1u << (b & 31));
    }
    __syncthreads();
  }
}

// ---------------- Edge features (zero-padded width to multiple of 4) ------
__global__ void edge_feat_kernel(const float* __restrict__ x, int C,
                                 const float* __restrict__ pos,
                                 const int* __restrict__ src, int sstride, int clampMax,
                                 int G, int t0, int numE,
                                 float* __restrict__ E)
{
  const int e = blockIdx.x;
  if (e >= numE) return;
  const int eg = t0 * G + e;
  int s = src[(size_t)eg * sstride];
  s = s < 0 ? 0 : (s > clampMax ? clampMax : s);
  const int t = t0 + e / G;
  const int Ct = C + 3;
  const int Ctp = (Ct + 3) & ~3;
  for (int c = threadIdx.x; c < Ctp; c += blockDim.x) {
    float v = 0.f;
    if (c < C)       v = x[(size_t)s * C + c];
    else if (c < Ct) v = pos[3*s + (c - C)] - pos[3*t + (c - C)];
    E[(size_t)e * Ctp + c] = v;
  }
}

// ---------------- weight pack: [K,N] -> pair-interleaved [Kp/2][N][2] ------
// (zero-fills rows K..Kp-1; gives the GEMM a single b64 load per B fragment)
__global__ void pack_w_kernel(const float* __restrict__ S, int K, int N, int Kp,
                              float* __restrict__ D)
{
  const int i = blockIdx.x * blockDim.x + threadIdx.x;   // pair index
  const int tot = (Kp >> 1) * N;
  if (i >= tot) return;
  const int kk = i / N, n = i - kk * N;
  const int k0 = 2 * kk, k1 = 2 * kk + 1;
  D[2*i]     = (k0 < K) ? S[(size_t)k0 * N + n] : 0.f;
  D[2*i + 1] = (k1 < K) ? S[(size_t)k1 * N + n] : 0.f;
}

// ---------------- activation column pad (pos -> [M,4]) ---------------------
__global__ void pad_cols_kernel(const float* __restrict__ S, int K, int Kp,
                                float* __restrict__ D, int M)
{
  const int i = blockIdx.x * blockDim.x + threadIdx.x;
  if (i >= M * Kp) return;
  const int r = i / Kp, c = i - r * Kp;
  D[i] = (c < K) ? S[(size_t)r * K + c] : 0.f;
}

// ---------------- WMMA f32 GEMM: 32x32 per wave, guard-free K loop ---------
// Requires: K % 4 == 0, lda == K, W pre-packed by pack_w_kernel.
// mode 0: D = A@W + bias
// mode 1: D = relu((A@W + bias) * g' + be)
// mode 2: D = (A@W + bias) * g' + be
__global__ void wmma_gemm_kernel(const float* __restrict__ A, int lda, int M, int K,
                                 const float* __restrict__ W,
                                 const float* __restrict__ bias,
                                 const float* __restrict__ bng,
                                 const float* __restrict__ bnb,
                                 float* __restrict__ D, int N, int mode)
{
  const int lane = threadIdx.x & 31;
  const int wvb  = threadIdx.x >> 5;                 // 4 waves / block
  const int row0 = (blockIdx.y * 4 + wvb) << 5;      // 32 rows / wave
  if (row0 >= M) return;                             // wave-uniform exit
  const int col0 = blockIdx.x << 5;                  // 32 cols / wave
  const int half = lane >> 4;
  const int l16  = lane & 15;
  // clamped fragment addresses: out-of-range rows/cols only feed outputs that
  // are masked at the store, so clamping is safe and keeps loads unconditional
  const int ar0 = (row0 + l16      < M) ? (row0 + l16)      : (M - 1);
  const int ar1 = (row0 + 16 + l16 < M) ? (row0 + 16 + l16) : (M - 1);
  const int bc0 = (col0 + l16      < N) ? (col0 + l16)      : (N - 1);
  const int bc1 = (col0 + 16 + l16 < N) ? (col0 + 16 + l16) : (N - 1);
  const float* A0 = A + (size_t)ar0 * lda + half * 2;
  const float* A1 = A + (size_t)ar1 * lda + half * 2;
  // packed W: pair row (k/2 + half), pair column bc -> float offset
  //           (k/2 + half)*2N + 2*bc  ==  k*N + half*2N + 2*bc
  const float* B0 = W + (size_t)half * 2 * N + (size_t)bc0 * 2;
  const float* B1 = W + (size_t)half * 2 * N + (size_t)bc1 * 2;
  __builtin_prefetch(A0, 0, 3);                      // global_prefetch_b8
  v8f acc00 = {0.f,0.f,0.f,0.f,0.f,0.f,0.f,0.f};
  v8f acc01 = acc00, acc10 = acc00, acc11 = acc00;
  for (int k = 0; k < K; k += 4) {
    const v2f a0 = *(const v2f*)(A0 + k);
    const v2f a1 = *(const v2f*)(A1 + k);
    const v2f b0 = *(const v2f*)(B0 + (size_t)k * N);
    const v2f b1 = *(const v2f*)(B1 + (size_t)k * N);
    acc00 = __builtin_amdgcn_wmma_f32_16x16x4_f32(false, a0, false, b0, (short)0, acc00, false, false);
    acc01 = __builtin_amdgcn_wmma_f32_16x16x4_f32(false, a0, false, b1, (short)0, acc01, false, false);
    acc10 = __builtin_amdgcn_wmma_f32_16x16x4_f32(false, a1, false, b0, (short)0, acc10, false, false);
    acc11 = __builtin_amdgcn_wmma_f32_16x16x4_f32(false, a1, false, b1, (short)0, acc11, false, false);
  }
#pragma unroll
  for (int j = 0; j < 2; ++j) {
    const int col = col0 + j * 16 + l16;
    if (col >= N) continue;
    const float bv = bias ? bias[col] : 0.f;
    float sc = 1.f, sh = 0.f;
    if (mode) { sc = bng[col] * rsqrtf(1.00001f); sh = bnb[col]; }
#pragma unroll
    for (int i2 = 0; i2 < 2; ++i2) {
      v8f acc = (j == 0) ? (i2 == 0 ? acc00 : acc10)
                         : (i2 == 0 ? acc01 : acc11);
#pragma unroll
      for (int r = 0; r < 8; ++r) {
        const int orow = row0 + i2 * 16 + r + half * 8;  // C/D rows r, r+8
        if (orow < M) {
          float v = acc[r] + bv;
          if (mode) v = v * sc + sh;
          if (mode == 1) v = fmaxf(v, 0.f);
          D[(size_t)orow * N + col] = v;
        }
      }
    }
  }
}

// ---------------- segment-max (contiguous groups) + BN + residual + ReLU ---
__global__ void segmax_merge_kernel(const float* __restrict__ conv, int G, int t0, int rows,
                                    const float* __restrict__ ident,
                                    const float* __restrict__ g, const float* __restrict__ b,
                                    float* __restrict__ y, int C)
{
  const int i = blockIdx.x * blockDim.x + threadIdx.x;
  if (i >= rows * C) return;
  const int tr = i / C, c = i - tr * C;
  const int t = t0 + tr;
  float v = 0.f;
  if (conv) {
    const float* p = conv + (size_t)tr * G * C + c;
    float m = -3.4e38f;
    for (int j = 0; j < G; ++j) m = fmaxf(m, p[(size_t)j * C]);
    v = m;
  }
  v = v * (g[c] * rsqrtf(1.00001f)) + b[c] + ident[(size_t)t * C + c];
  y[(size_t)t * C + c] = fmaxf(v, 0.f);
}

// ---------------- row gather ----------------------------------------------
__global__ void gather_rows_kernel(const float* __restrict__ S, const int* __restrict__ idx,
                                   float* __restrict__ D, int M, int C)
{
  const int i = blockIdx.x * blockDim.x + threadIdx.x;
  if (i >= M * C) return;
  const int r = i / C, c = i - r * C;
  D[i] = S[(size_t)idx[r] * C + c];
}

// ---------------- column concat into padded width --------------------------
__global__ void concat_kernel(const float* __restrict__ A, int Ca,
                              const float* __restrict__ B, int Cb,
                              float* __restrict__ D, int M, int Cp)
{
  const int i = blockIdx.x * blockDim.x + threadIdx.x;
  if (i >= M * Cp) return;
  const int r = i / Cp, c = i - r * Cp;
  float v = 0.f;
  if (c < Ca)           v = A[(size_t)r * Ca + c];
  else if (c < Ca + Cb) v = B[(size_t)r * Cb + (c - Ca)];
  D[i] = v;
}

// ---------------- kNN interpolate (k=3), one wave per target ---------------
__global__ void knn_interp_kernel(const float* __restrict__ X, const float* __restrict__ psrc,
                                  const float* __restrict__ ptgt, float* __restrict__ out,
                                  int Ns, int Nt, int C)
{
  const int gt = blockIdx.x * blockDim.x + threadIdx.x;
  const int wv = gt >> 5, lane = gt & 31;
  if (wv >= Nt) return;
  const float tx = ptgt[3*wv], ty = ptgt[3*wv+1], tz = ptgt[3*wv+2];
  int ch[3] = {-1, -1, -1};
  float cd[3] = {0.f, 0.f, 0.f};
#pragma unroll
  for (int k = 0; k < 3; ++k) {
    float bm = 3.4e38f; int bi = 0x7fffffff;
    for (int j = lane; j < Ns; j += 32) {
      bool skip = false;
#pragma unroll
      for (int q = 0; q < 3; ++q) if (q < k && ch[q] == j) skip = true;
      if (skip) continue;
      float dx = psrc[3*j]-tx, dy = psrc[3*j+1]-ty, dz = psrc[3*j+2]-tz;
      float d = dx*dx + dy*dy + dz*dz;
      if (d < bm || (d == bm && j < bi)) { bm = d; bi = j; }
    }
#pragma unroll
    for (int o = 16; o; o >>= 1) {
      float om = __shfl_xor(bm, o, 32);
      int   oi = __shfl_xor(bi, o, 32);
      if (om < bm || (om == bm && oi < bi)) { bm = om; bi = oi; }
    }
    ch[k] = bi; cd[k] = bm;
  }
  float w0 = 1.f / (fmaxf(cd[0], 0.f) + 1e-16f);
  float w1 = 1.f / (fmaxf(cd[1], 0.f) + 1e-16f);
  float w2 = 1.f / (fmaxf(cd[2], 0.f) + 1e-16f);
  const float iws = 1.f / (w0 + w1 + w2);
  w0 *= iws; w1 *= iws; w2 *= iws;
  const float* x0 = X + (size_t)ch[0] * C;
  const float* x1 = X + (size_t)ch[1] * C;
  const float* x2 = X + (size_t)ch[2] * C;
  for (int c = lane; c < C; c += 32)
    out[(size_t)wv * C + c] = w0 * x0[c] + w1 * x1[c] + w2 * x2[c];
}

// ---------------- ArcFace head: scaled cosine ------------------------------
__global__ void arc_head_kernel(const float* __restrict__ emb, const float* __restrict__ w,
                                float* __restrict__ out, int N)
{
  const int gt = blockIdx.x * blockDim.x + threadIdx.x;
  const int wv = gt >> 5, lane = gt & 31;
  if (wv >= N) return;
  const float* e = emb + (size_t)wv * 128;
  const float e0 = e[lane], e1 = e[lane+32], e2 = e[lane+64], e3 = e[lane+96];
  float ss = e0*e0 + e1*e1 + e2*e2 + e3*e3;
#pragma unroll
  for (int o = 16; o; o >>= 1) ss += __shfl_xor(ss, o, 32);
  const float inv_e = rsqrtf(ss);
#pragma unroll
  for (int c = 0; c < 3; ++c) {
    const float* wr = w + c * 128;
    const float w0 = wr[lane], w1 = wr[lane+32], w2 = wr[lane+64], w3 = wr[lane+96];
    float dt  = e0*w0 + e1*w1 + e2*w2 + e3*w3;
    float wss = w0*w0 + w1*w1 + w2*w2 + w3*w3;
#pragma unroll
    for (int o = 16; o; o >>= 1) { dt += __shfl_xor(dt, o, 32); wss += __shfl_xor(wss, o, 32); }
    if (lane == 0) out[(size_t)wv * 3 + c] = 30.0f * dt * inv_e * rsqrtf(wss);
  }
}

// ===========================================================================
// Host orchestration
// ===========================================================================
struct Blk { const float *l1W,*l1b,*g1,*be1,*l2W,*l2b,*bng,*bnb,*sW,*sb,*sg,*sbe; };
struct Mlp { const float *l1W,*l1b,*g1,*be1,*l2W,*l2b; };

static void gemm(hipStream_t st, const float* A, int lda, int M, int K,
                 const float* W, const float* bias, const float* g, const float* b,
                 float* D, int N, int mode)
{
  dim3 grid((unsigned)((N + 31) / 32), (unsigned)((M + 127) / 128), 1);
  wmma_gemm_kernel<<<grid, 128, 0, st>>>(A, lda, M, K, W, bias, g, b, D, N, mode);
}

// x: node features for edge gather / residual identity; xA/Ka: (padded) GEMM
// operand for the shortcut branch.
static void run_block(hipStream_t st, const float* x, int C,
                      const float* xA, int Ka,
                      const float* pos, int Np,
                      const int* src, int sstride, int G, int Mt, int Nn,
                      const Blk& P, int H, float* identBuf,
                      float* Ebuf, float* Hbuf, float* Lbuf, float* y)
{
  const float* ident = x;                        // cin == cout case
  if (P.sW) {
    gemm(st, xA, Ka, Nn, Ka, P.sW, P.sb, P.sg, P.sbe, identBuf, H, 2);
    ident = identBuf;
  }
  const int Cin = C + 3;
  const int Cp  = (Cin + 3) & ~3;                // padded K for layer-1 GEMM
  const int TC  = 1024;                          // target chunk (bounds scratch)
  for (int t0 = 0; t0 < Mt; t0 += TC) {
    int mt = Mt - t0; if (mt > TC) mt = TC;
    const int nE = mt * G;
    edge_feat_kernel<<<nE, 64, 0, st>>>(x, C, pos, src, sstride, Np - 1, G, t0, nE, Ebuf);
    gemm(st, Ebuf, Cp, nE, Cp, P.l1W, P.l1b, P.g1, P.be1, Hbuf, H, 1);
    gemm(st, Hbuf, H, nE, H, P.l2W, P.l2b, nullptr, nullptr, Lbuf, H, 0);
    const int tot = mt * H;
    segmax_merge_kernel<<<(tot + 255)/256, 256, 0, st>>>(Lbuf, G, t0, mt, ident,
                                                         P.bng, P.bnb, y, H);
  }
  if (Nn > Mt) {                                 // nodes without edges: conv = 0
    const int tot = (Nn - Mt) * H;
    segmax_merge_kernel<<<(tot + 255)/256, 256, 0, st>>>(nullptr, G, Mt, Nn - Mt, ident,
                                                         P.bng, P.bnb, y, H);
  }
}

static void run_mlp(hipStream_t st, const float* in, int M, int K, const Mlp& P,
                    int Hd, int Od, float* hbuf, float* out)
{
  gemm(st, in, K, M, K, P.l1W, P.l1b, P.g1, P.be1, hbuf, Hd, 1);
  gemm(st, hbuf, Hd, M, Hd, P.l2W, P.l2b, nullptr, nullptr, out, Od, 0);
}

extern "C" void kernel_launch(void* const* d_in, const int* in_sizes, int n_in,
                              void* d_out, int out_size, void* d_ws, size_t ws_size,
                              hipStream_t stream)
{
  (void)in_sizes; (void)n_in; (void)out_size; (void)ws_size;
  const float* pos = (const float*)d_in[0];
  int p = 1;
  auto F = [&]() { return (const float*)d_in[p++]; };

  Blk e1;  e1.l1W=F();  e1.l1b=F();  e1.g1=F();  e1.be1=F();  e1.l2W=F();  e1.l2b=F();
           e1.bng=F();  e1.bnb=F();  e1.sW=F();  e1.sb=F();   e1.sg=F();   e1.sbe=F();
  Blk e1b; e1b.l1W=F(); e1b.l1b=F(); e1b.g1=F(); e1b.be1=F(); e1b.l2W=F(); e1b.l2b=F();
           e1b.bng=F(); e1b.bnb=F(); e1b.sW=nullptr; e1b.sb=nullptr; e1b.sg=nullptr; e1b.sbe=nullptr;
  Blk e2;  e2.l1W=F();  e2.l1b=F();  e2.g1=F();  e2.be1=F();  e2.l2W=F();  e2.l2b=F();
           e2.bng=F();  e2.bnb=F();  e2.sW=F();  e2.sb=F();   e2.sg=F();   e2.sbe=F();
  Blk e2b; e2b.l1W=F(); e2b.l1b=F(); e2b.g1=F(); e2b.be1=F(); e2b.l2W=F(); e2b.l2b=F();
           e2b.bng=F(); e2b.bnb=F(); e2b.sW=nullptr; e2b.sb=nullptr; e2b.sg=nullptr; e2b.sbe=nullptr;
  Blk e3;  e3.l1W=F();  e3.l1b=F();  e3.g1=F();  e3.be1=F();  e3.l2W=F();  e3.l2b=F();
           e3.bng=F();  e3.bnb=F();  e3.sW=F();  e3.sb=F();   e3.sg=F();   e3.sbe=F();
  Mlp f3;  f3.l1W=F();  f3.l1b=F();  f3.g1=F();  f3.be1=F();  f3.l2W=F();  f3.l2b=F();
  Mlp f2;  f2.l1W=F();  f2.l1b=F();  f2.g1=F();  f2.be1=F();  f2.l2W=F();  f2.l2b=F();
  Mlp f1;  f1.l1W=F();  f1.l1b=F();  f1.g1=F();  f1.be1=F();  f1.l2W=F();  f1.l2b=F();
  const float* arcw = F();

  // ---- workspace bump allocator (float granularity, 256B aligned) ----
  float* fw = (float*)d_ws;
  size_t off = 0;
  auto fal = [&](size_t n) { float* q = fw + off; off += (n + 63) & ~(size_t)63; return q; };

  float* dist   = fal(NN0);
  float* pos1   = fal(NN1 * 3);
  float* pos2   = fal(NN2 * 3);
  float* pos3   = fal(NN3 * 3);
  float* posP   = fal(NN0 * 4);     // zero-padded pos for enc1 shortcut GEMM
  float* x1a    = fal(NN1 * 64);
  float* x1     = fal(NN1 * 64);
  float* x2a    = fal(NN2 * 128);
  float* x2     = fal(NN2 * 128);
  float* x3     = fal(NN3 * 256);
  float* identB = fal(NN0 * 64);
  float* yB     = fal(NN0 * 64);
  float* Ebuf   = fal(32768 * 68);
  float* Hbuf   = fal(32768 * 128);
  float* Lbuf   = fal(32768 * 128);
  float* upB    = fal(NN0 * 128);
  float* catB   = fal(NN0 * 132);
  float* fhB    = fal(NN0 * 128);
  float* xu2    = fal(NN2 * 128);
  float* xu1    = fal(NN1 * 128);
  float* emb    = fal(NN0 * 128);
  int* idx1 = (int*)fal(NN1);
  int* idx2 = (int*)fal(NN2);
  int* idx3 = (int*)fal(NN3);
  int* src1 = (int*)fal(NN1 * 32);
  int* src2 = (int*)fal(NN2 * 32);
  int* src3 = (int*)fal(NN3 * 16);

  hipStream_t st = stream;
  int tot;

  // ---- pack ALL GEMM weights: zero-pad K to x4 + pair-interleave rows ----
  auto prepW = [&](const float*& Wp, int K, int N) {
    const int Kp = (K + 3) & ~3;
    float* dst = fal((size_t)Kp * N);
    const int tot2 = (Kp >> 1) * N;
    pack_w_kernel<<<(tot2 + 255)/256, 256, 0, st>>>(Wp, K, N, Kp, dst);
    Wp = dst;
  };
  prepW(e1.l1W,   6,  64);  prepW(e1.l2W,  64,  64);  prepW(e1.sW,    3,  64);
  prepW(e1b.l1W, 67,  64);  prepW(e1b.l2W, 64,  64);
  prepW(e2.l1W,  67, 128);  prepW(e2.l2W, 128, 128);  prepW(e2.sW,   64, 128);
  prepW(e2b.l1W,131, 128);  prepW(e2b.l2W,128, 128);
  prepW(e3.l1W, 131, 256);  prepW(e3.l2W, 256, 256);  prepW(e3.sW,  128, 256);
  prepW(f3.l1W, 384, 256);  prepW(f3.l2W, 256, 128);
  prepW(f2.l1W, 192, 128);  prepW(f2.l2W, 128, 128);
  prepW(f1.l1W, 131, 128);  prepW(f1.l2W, 128, 128);
  pad_cols_kernel<<<(NN0*4 + 255)/256, 256, 0, st>>>(pos, 3, 4, posP, NN0);

  // ======== encoder level 1 ========
  fps_kernel<<<1, 256, 0, st>>>(pos, NN0, NN1, idx1, dist);
  knn_kernel<<<NN1, 256, 0, st>>>(pos, idx1, NN0, 32, src1);
  run_block(st, pos, 3, posP, 4, pos, NN0, src1, 1, 32, NN1, NN0, e1, 64,
            identB, Ebuf, Hbuf, Lbuf, yB);
  tot = NN1 * 64; gather_rows_kernel<<<(tot+255)/256, 256, 0, st>>>(yB, idx1, x1a, NN1, 64);
  tot = NN1 * 3;  gather_rows_kernel<<<(tot+255)/256, 256, 0, st>>>(pos, idx1, pos1, NN1, 3);
  run_block(st, x1a, 64, x1a, 64, pos1, NN1, src1, 4, 8, NN1, NN1, e1b, 64,
            identB, Ebuf, Hbuf, Lbuf, x1);

  // ======== encoder level 2 ========
  fps_kernel<<<1, 256, 0, st>>>(pos1, NN1, NN2, idx2, dist);
  knn_kernel<<<NN2, 256, 0, st>>>(pos1, idx2, NN1, 32, src2);
  run_block(st, x1, 64, x1, 64, pos1, NN1, src2, 1, 32, NN2, NN1, e2, 128,
            identB, Ebuf, Hbuf, Lbuf, yB);
  tot = NN2 * 128; gather_rows_kernel<<<(tot+255)/256, 256, 0, st>>>(yB, idx2, x2a, NN2, 128);
  tot = NN2 * 3;   gather_rows_kernel<<<(tot+255)/256, 256, 0, st>>>(pos1, idx2, pos2, NN2, 3);
  run_block(st, x2a, 128, x2a, 128, pos2, NN2, src2, 4, 8, NN2, NN2, e2b, 128,
            identB, Ebuf, Hbuf, Lbuf, x2);

  // ======== encoder level 3 ========
  fps_kernel<<<1, 256, 0, st>>>(pos2, NN2, NN3, idx3, dist);
  knn_kernel<<<NN3, 256, 0, st>>>(pos2, idx3, NN2, 16, src3);
  run_block(st, x2, 128, x2, 128, pos2, NN2, src3, 1, 16, NN3, NN2, e3, 256,
            identB, Ebuf, Hbuf, Lbuf, yB);
  tot = NN3 * 256; gather_rows_kernel<<<(tot+255)/256, 256, 0, st>>>(yB, idx3, x3, NN3, 256);
  tot = NN3 * 3;   gather_rows_kernel<<<(tot+255)/256, 256, 0, st>>>(pos2, idx3, pos3, NN3, 3);

  // ======== feature propagation ========
  knn_interp_kernel<<<(NN2 + 7)/8, 256, 0, st>>>(x3, pos3, pos2, upB, NN3, NN2, 256);
  tot = NN2 * 384; concat_kernel<<<(tot+255)/256, 256, 0, st>>>(upB, 256, x2, 128, catB, NN2, 384);
  run_mlp(st, catB, NN2, 384, f3, 256, 128, fhB, xu2);

  knn_interp_kernel<<<(NN1 + 7)/8, 256, 0, st>>>(xu2, pos2, pos1, upB, NN2, NN1, 128);
  tot = NN1 * 192; concat_kernel<<<(tot+255)/256, 256, 0, st>>>(upB, 128, x1, 64, catB, NN1, 192);
  run_mlp(st, catB, NN1, 192, f2, 128, 128, fhB, xu1);

  knn_interp_kernel<<<(NN0 + 7)/8, 256, 0, st>>>(xu1, pos1, pos, upB, NN1, NN0, 128);
  tot = NN0 * 132; concat_kernel<<<(tot+255)/256, 256, 0, st>>>(upB, 128, pos, 3, catB, NN0, 132);
  run_mlp(st, catB, NN0, 132, f1, 128, 128, fhB, emb);

  // ======== ArcFace head ========
  arc_head_kernel<<<NN0/8, 256, 0, st>>>(emb, arcw, (float*)d_out, NN0);
}